// GCNConvModel_78546361909653
// MI455X (gfx1250) — compile-verified
//
#include <hip/hip_runtime.h>
#include <hip/hip_bf16.h>

// ---------------------------------------------------------------------------
// GCN model on MI455X (gfx1250, wave32).
// Dense GEMMs use V_WMMA_F32_16X16X4_F32 (exact fp32, D = A(16x4) * B(4x16) + C).
// Roofline: ~2.3 GFLOP dense vs ~50M f32 atomics + 220MB/layer gather, all
// L2-resident (192MB L2) -> scatter/atomic bound; fp32 WMMA costs nothing and
// matches reference numerics exactly.
// ---------------------------------------------------------------------------

typedef float v2f __attribute__((ext_vector_type(2)));
typedef float v8f __attribute__((ext_vector_type(8)));

#define N_NODES   50000
#define N_EDGES   800000
#define IN_CH     128
#define HID       64
#define NEG_SLOPE 0.2f

// ---------------------------------------------------------------------------
// WMMA fp32 GEMM: Y[nrows x NOUT] = epilogue(X[nrows x K] @ W[K x NOUT])
// One wave computes a 16 x NOUT tile; 8 waves per 256-thread block.
// W is staged TRANSPOSED in LDS (pitch K+4):
//   - B fragment (2 consecutive K-values of one column) becomes one ds_load_b64
//   - pitch K+4 -> 272B column stride -> bank step of 4 -> lanes 0..15 hit 16
//     distinct banks (conflict-free fragment reads)
//
// wave32 fragment layouts (CDNA5 ISA 7.12.2):
//   A 16x4 f32 : lane half h=(lane>>4), elem j: M = lane&15, K = 2h + j
//   B 4x16 f32 : lane half h,            elem j: K = 2h + j, N = lane&15
//   C 16x16 f32: vgpr r:                 M = r + 8h,         N = lane&15
// ---------------------------------------------------------------------------
template <int K, int NOUT, bool ADD_BIAS, bool RELU>
__global__ __launch_bounds__(256) void gemm_wmma_f32(
    const float* __restrict__ X, const float* __restrict__ W,
    const float* __restrict__ bias, float* __restrict__ Y, int nrows) {
  constexpr int PITCH = K + 4;
  __shared__ __align__(16) float sWt[NOUT * PITCH];  // transposed: [n][k]
  for (int idx = threadIdx.x; idx < K * NOUT; idx += blockDim.x) {
    const int kk = idx / NOUT;   // coalesced global read (K-major in memory)
    const int nn = idx % NOUT;
    sWt[nn * PITCH + kk] = W[idx];
  }
  __syncthreads();

  const int wave = threadIdx.x >> 5;
  const int lane = threadIdx.x & 31;
  const int tile = blockIdx.x * 8 + wave;
  if (tile * 16 >= nrows) return;  // uniform per wave -> EXEC stays all-1s

  const int m0   = tile * 16;
  const int half = lane >> 4;   // 0 or 1
  const int ln16 = lane & 15;
  const int kh   = half * 2;    // K-offset of this lane's fragment elements
  const int row  = m0 + ln16;
  const float* __restrict__ xrow = X + (size_t)row * K;

  // warm the strided per-lane A row (global_prefetch_b8)
  __builtin_prefetch(xrow + K / 2, 0, 1);

  constexpr int NT = NOUT / 16;
  v8f acc[NT];
#pragma unroll
  for (int t = 0; t < NT; ++t)
#pragma unroll
    for (int r = 0; r < 8; ++r) acc[t][r] = 0.0f;

#pragma unroll 4
  for (int k = 0; k < K; k += 4) {
    v2f a;
    a.x = xrow[k + kh + 0];
    a.y = xrow[k + kh + 1];     // merged into one global_load_b64
#pragma unroll
    for (int t = 0; t < NT; ++t) {
      // contiguous + 8B aligned: (t*16+ln16)*PITCH + k + kh is even
      const v2f b = *(const v2f*)&sWt[(t * 16 + ln16) * PITCH + k + kh];
      acc[t] = __builtin_amdgcn_wmma_f32_16x16x4_f32(
          /*neg_a=*/false, a, /*neg_b=*/false, b,
          /*c_mod=*/(short)0, acc[t], /*reuse_a=*/false, /*reuse_b=*/false);
    }
  }

#pragma unroll
  for (int t = 0; t < NT; ++t) {
    const int col = t * 16 + ln16;
    const float bv = ADD_BIAS ? bias[col] : 0.0f;
#pragma unroll
    for (int r = 0; r < 8; ++r) {
      float v = acc[t][r] + bv;
      if (RELU) v = fmaxf(v, 0.0f);
      Y[(m0 + r + half * 8) * NOUT + col] = v;
    }
  }
}

// ---------------------------------------------------------------------------
// Degree / normalization
// ---------------------------------------------------------------------------
__global__ void deg_count_kernel(const int* __restrict__ dst,
                                 float* __restrict__ deg, int E) {
  int e = blockIdx.x * blockDim.x + threadIdx.x;
  if (e < E) atomicAdd(&deg[dst[e]], 1.0f);
}

__global__ void deg_isqrt_kernel(const float* __restrict__ deg,
                                 float* __restrict__ disq, int N) {
  int i = blockIdx.x * blockDim.x + threadIdx.x;
  if (i < N) disq[i] = rsqrtf(deg[i] + 1.0f);  // +1 = self loop; deg_total >= 1
}

// ---------------------------------------------------------------------------
// Edge scatter: ONE WAVE PER EDGE. Lane L handles channels {2L, 2L+1}:
//   - src/dst/disq loaded once per wave (same address across lanes -> one line)
//   - h gather is a coalesced float2 (b64) load of one 256B row
//   - 2x global_atomic_add_f32 into L2-resident agg
// ---------------------------------------------------------------------------
__global__ __launch_bounds__(256) void edge_scatter_kernel(
    const int* __restrict__ src, const int* __restrict__ dst,
    const float* __restrict__ disq, const float* __restrict__ h,
    float* __restrict__ agg, int E) {
  const int wave = threadIdx.x >> 5;
  const int lane = threadIdx.x & 31;
  const int e = blockIdx.x * 8 + wave;
  if (e < E) {
    const int s = src[e];
    const int d = dst[e];
    const float nrm = disq[s] * disq[d];
    const float2 hv = *(const float2*)&h[s * HID + lane * 2];
    float* ap = &agg[d * HID + lane * 2];
    atomicAdd(ap + 0, hv.x * nrm);
    atomicAdd(ap + 1, hv.y * nrm);
  }
}

// ---------------------------------------------------------------------------
// Finalize (float4 lanes): out = act(agg + h*disq^2 + bias) [+ residual]
// (self-loop contribution h[i]*disq[i]^2 folded in here, after scatter)
// ---------------------------------------------------------------------------
template <bool LRELU, bool RES>
__global__ __launch_bounds__(256) void finalize_kernel(
    const float* __restrict__ agg, const float* __restrict__ h,
    const float* __restrict__ disq, const float* __restrict__ bias,
    const float* __restrict__ res, float* __restrict__ out, int N) {
  const int t = blockIdx.x * blockDim.x + threadIdx.x;  // one float4 per thread
  if (t < N * (HID / 4)) {
    const int i  = t >> 4;        // node
    const int c4 = t & 15;        // float4 chunk within row
    const float dd = disq[i] * disq[i];
    const float4 av = ((const float4*)agg)[t];
    const float4 hv = ((const float4*)h)[t];
    const float4 bv = ((const float4*)bias)[c4];
    float4 v;
    v.x = av.x + hv.x * dd + bv.x;
    v.y = av.y + hv.y * dd + bv.y;
    v.z = av.z + hv.z * dd + bv.z;
    v.w = av.w + hv.w * dd + bv.w;
    if (LRELU) {
      v.x = (v.x >= 0.0f) ? v.x : NEG_SLOPE * v.x;
      v.y = (v.y >= 0.0f) ? v.y : NEG_SLOPE * v.y;
      v.z = (v.z >= 0.0f) ? v.z : NEG_SLOPE * v.z;
      v.w = (v.w >= 0.0f) ? v.w : NEG_SLOPE * v.w;
    }
    if (RES) {
      const float4 rv = ((const float4*)res)[t];
      v.x += rv.x; v.y += rv.y; v.z += rv.z; v.w += rv.w;
    }
    ((float4*)out)[t] = v;
  }
}

// ---------------------------------------------------------------------------
// Head: out[N x 2] = X[N x 64] @ Wf[64 x 2] + bf
// ---------------------------------------------------------------------------
__global__ __launch_bounds__(256) void final_linear_kernel(
    const float* __restrict__ X, const float* __restrict__ W,
    const float* __restrict__ b, float* __restrict__ out, int N) {
  int i = blockIdx.x * blockDim.x + threadIdx.x;
  if (i < N) {
    float a0 = b[0], a1 = b[1];
#pragma unroll
    for (int k = 0; k < HID; k += 2) {
      const float4 wv = *(const float4*)&W[k * 2];      // W[k][0..1], W[k+1][0..1]
      const float2 xv = *(const float2*)&X[i * HID + k];
      a0 += xv.x * wv.x + xv.y * wv.z;
      a1 += xv.x * wv.y + xv.y * wv.w;
    }
    out[i * 2 + 0] = a0;
    out[i * 2 + 1] = a1;
  }
}

// ---------------------------------------------------------------------------
extern "C" void kernel_launch(void* const* d_in, const int* in_sizes, int n_in,
                              void* d_out, int out_size, void* d_ws,
                              size_t ws_size, hipStream_t stream) {
  const float* nf  = (const float*)d_in[0];
  const int*   ei  = (const int*)d_in[1];
  const float* W1  = (const float*)d_in[2];
  const float* b1  = (const float*)d_in[3];
  const float* W2  = (const float*)d_in[4];
  const float* b2  = (const float*)d_in[5];
  const float* Wc1 = (const float*)d_in[6];
  const float* bc1 = (const float*)d_in[7];
  const float* Wc2 = (const float*)d_in[8];
  const float* bc2 = (const float*)d_in[9];
  const float* Wc3 = (const float*)d_in[10];
  const float* bc3 = (const float*)d_in[11];
  const float* Wc4 = (const float*)d_in[12];
  const float* bc4 = (const float*)d_in[13];
  const float* Wf  = (const float*)d_in[14];
  const float* bf  = (const float*)d_in[15];
  float* out = (float*)d_out;

  const int N = N_NODES;
  const int E = N_EDGES;
  const int* src = ei;
  const int* dst = ei + E;

  // Workspace layout (~71 MB of f32)
  float* ws = (float*)d_ws;
  size_t off = 0;
  float* B0   = ws + off; off += (size_t)N * HID;  // x, later out3
  float* Bh   = ws + off; off += (size_t)N * HID;  // h = x @ Wc
  float* Ba   = ws + off; off += (size_t)N * HID;  // scatter accumulator
  float* B3   = ws + off; off += (size_t)N * HID;  // out1, later out4
  float* B4   = ws + off; off += (size_t)N * HID;  // out2
  float* tmp  = ws + off; off += (size_t)N * 32;   // fc1 activations
  float* deg  = ws + off; off += (size_t)N;
  float* disq = ws + off; off += (size_t)N;

  const int BLK = 256;
  const int gemm_blocks = (N / 16 + 7) / 8;          // 3125 tiles, 8 waves/blk
  const int scat_blocks = (E + 7) / 8;               // 1 wave / edge
  const int fin_blocks  = (N * (HID / 4) + BLK - 1) / BLK;
  const size_t aggB = (size_t)N * HID * sizeof(float);

  // --- degrees & symmetric normalization -----------------------------------
  hipMemsetAsync(deg, 0, (size_t)N * sizeof(float), stream);
  deg_count_kernel<<<(E + BLK - 1) / BLK, BLK, 0, stream>>>(dst, deg, E);
  deg_isqrt_kernel<<<(N + BLK - 1) / BLK, BLK, 0, stream>>>(deg, disq, N);

  // --- fc stack: x = relu(nf @ W1 + b1) @ W2 + b2 --------------------------
  gemm_wmma_f32<IN_CH, 32, true, true><<<gemm_blocks, BLK, 0, stream>>>(nf, W1, b1, tmp, N);
  gemm_wmma_f32<32, HID, true, false><<<gemm_blocks, BLK, 0, stream>>>(tmp, W2, b2, B0, N);

  // --- conv layer 1: out1 = lrelu(gcn(x)) ----------------------------------
  gemm_wmma_f32<HID, HID, false, false><<<gemm_blocks, BLK, 0, stream>>>(B0, Wc1, nullptr, Bh, N);
  hipMemsetAsync(Ba, 0, aggB, stream);
  edge_scatter_kernel<<<scat_blocks, BLK, 0, stream>>>(src, dst, disq, Bh, Ba, E);
  finalize_kernel<true, false><<<fin_blocks, BLK, 0, stream>>>(Ba, Bh, disq, bc1, nullptr, B3, N);

  // --- conv layer 2: out2 = lrelu(gcn(out1)) + out1 ------------------------
  gemm_wmma_f32<HID, HID, false, false><<<gemm_blocks, BLK, 0, stream>>>(B3, Wc2, nullptr, Bh, N);
  hipMemsetAsync(Ba, 0, aggB, stream);
  edge_scatter_kernel<<<scat_blocks, BLK, 0, stream>>>(src, dst, disq, Bh, Ba, E);
  finalize_kernel<true, true><<<fin_blocks, BLK, 0, stream>>>(Ba, Bh, disq, bc2, B3, B4, N);

  // --- conv layer 3: out3 = lrelu(gcn(out2)) -------------------------------
  gemm_wmma_f32<HID, HID, false, false><<<gemm_blocks, BLK, 0, stream>>>(B4, Wc3, nullptr, Bh, N);
  hipMemsetAsync(Ba, 0, aggB, stream);
  edge_scatter_kernel<<<scat_blocks, BLK, 0, stream>>>(src, dst, disq, Bh, Ba, E);
  finalize_kernel<true, false><<<fin_blocks, BLK, 0, stream>>>(Ba, Bh, disq, bc3, nullptr, B0, N);

  // --- conv layer 4: out4 = gcn(out3) + out2 (no activation) ---------------
  gemm_wmma_f32<HID, HID, false, false><<<gemm_blocks, BLK, 0, stream>>>(B0, Wc4, nullptr, Bh, N);
  hipMemsetAsync(Ba, 0, aggB, stream);
  edge_scatter_kernel<<<scat_blocks, BLK, 0, stream>>>(src, dst, disq, Bh, Ba, E);
  finalize_kernel<false, true><<<fin_blocks, BLK, 0, stream>>>(Ba, Bh, disq, bc4, B4, B3, N);

  // --- head ----------------------------------------------------------------
  final_linear_kernel<<<(N + BLK - 1) / BLK, BLK, 0, stream>>>(B3, Wf, bf, out, N);
}